// PAdicRankingLossV2_13305808683582
// MI455X (gfx1250) — compile-verified
//
#include <hip/hip_runtime.h>
#include <hip/hip_bf16.h>
#include <math.h>

#define B_SZ      16384
#define D_SZ      64
#define NCAND     1000
#define NCAND_PAD 1008
#define NPOS      3
#define NHALF     8191
#define NNEG      8192
#define NHARD     250
#define NRAND     250
#define MRAND     500
#define NTRIP     500

typedef float v2f __attribute__((ext_vector_type(2)));
typedef float v8f __attribute__((ext_vector_type(8)));

// ---------------------------------------------------------------- utilities

__device__ __forceinline__ int iabs_i(int x) { return x < 0 ? -x : x; }

// valuation: sum_{k=1..9} [d % 3^k == 0]  (d==0 -> 9)
__device__ __forceinline__ int val9(int d) {
  if (d == 0) return 9;
  int v = 0;
  while (v < 9 && (d % 3) == 0) { d /= 3; ++v; }
  return v;
}

__device__ __forceinline__ unsigned rotl32(unsigned x, int r) {
  return (x << r) | (x >> (32 - r));
}

// Threefry-2x32, 20 rounds, matching jax.random's schedule exactly.
__device__ void threefry2x32(unsigned k0, unsigned k1, unsigned c0, unsigned c1,
                             unsigned& o0, unsigned& o1) {
  unsigned ks0 = k0, ks1 = k1, ks2 = 0x1BD11BDAu ^ k0 ^ k1;
  unsigned x0 = c0 + ks0, x1 = c1 + ks1;
#define TF_R(r) { x0 += x1; x1 = rotl32(x1, r); x1 ^= x0; }
#define TF_GA() TF_R(13) TF_R(15) TF_R(26) TF_R(6)
#define TF_GB() TF_R(17) TF_R(29) TF_R(16) TF_R(24)
  TF_GA(); x0 += ks1; x1 += ks2 + 1u;
  TF_GB(); x0 += ks2; x1 += ks0 + 2u;
  TF_GA(); x0 += ks0; x1 += ks1 + 3u;
  TF_GB(); x0 += ks1; x1 += ks2 + 4u;
  TF_GA(); x0 += ks2; x1 += ks0 + 5u;
#undef TF_GA
#undef TF_GB
#undef TF_R
  o0 = x0; o1 = x1;
}

__device__ __forceinline__ bool hard_cond(float dan, float dap, float vp, float vn) {
  float vd = vp - vn;
  if (!(vd > 0.0f)) return false;
  float thr = (dap + 0.05f) + 0.15f * vd;   // d_ap + BASE, then + SCALE*vdiff
  return dan < thr;
}

// ------------------------------------------------------------ K0: rng setup
// Reproduce: ka,kr = split(key(42)); anchors = randint(ka,(1000,),0,B)
// k1,k2,k3 = split(kr,3); ra,rp,rn = randint(k*,(500,),0,B); distinct filter.
__global__ void rng_kernel(const int* __restrict__ bidx,
                           int* __restrict__ anchors,
                           int* __restrict__ rab, int* __restrict__ rpb, int* __restrict__ rnb,
                           int* __restrict__ r_a, int* __restrict__ r_p, int* __restrict__ r_n,
                           float* __restrict__ r_vp, float* __restrict__ r_vn,
                           float* __restrict__ r_mask,
                           int* __restrict__ h_a, int* __restrict__ h_p, int* __restrict__ h_n,
                           float* __restrict__ h_vp, float* __restrict__ h_vn) {
  int t = threadIdx.x;
  // split(key(42), 2): counts [0,1,2,3] -> pairs (0,2) and (1,3)
  unsigned a0, a1, b0, b1;
  threefry2x32(0u, 42u, 0u, 2u, a0, b0);
  threefry2x32(0u, 42u, 1u, 3u, a1, b1);
  unsigned ka0 = a0, ka1 = a1, kr0 = b0, kr1 = b1;
  // split(kr, 3): counts [0..5] -> pairs (0,3),(1,4),(2,5); bits=[s0,s1,s2,s3,s4,s5]
  unsigned s0, s1, s2, s3, s4, s5;
  threefry2x32(kr0, kr1, 0u, 3u, s0, s3);
  threefry2x32(kr0, kr1, 1u, 4u, s1, s4);
  threefry2x32(kr0, kr1, 2u, 5u, s2, s5);
  // k1=(s0,s1) k2=(s2,s3) k3=(s4,s5)

  if (t < 500) {   // anchors: random_bits pairs (i, i+500)
    unsigned o0, o1;
    threefry2x32(ka0, ka1, (unsigned)t, (unsigned)(t + 500), o0, o1);
    anchors[t]       = (int)(o0 % (unsigned)B_SZ);
    anchors[t + 500] = (int)(o1 % (unsigned)B_SZ);
  }
  if (t < 250) {   // 500 draws each: pairs (i, i+250)
    unsigned o0, o1;
    threefry2x32(s0, s1, (unsigned)t, (unsigned)(t + 250), o0, o1);
    rab[t] = (int)(o0 % (unsigned)B_SZ); rab[t + 250] = (int)(o1 % (unsigned)B_SZ);
    threefry2x32(s2, s3, (unsigned)t, (unsigned)(t + 250), o0, o1);
    rpb[t] = (int)(o0 % (unsigned)B_SZ); rpb[t + 250] = (int)(o1 % (unsigned)B_SZ);
    threefry2x32(s4, s5, (unsigned)t, (unsigned)(t + 250), o0, o1);
    rnb[t] = (int)(o0 % (unsigned)B_SZ); rnb[t + 250] = (int)(o1 % (unsigned)B_SZ);
  }
  if (t < NHARD) { h_a[t] = 0; h_p[t] = 0; h_n[t] = 0; h_vp[t] = 0.0f; h_vn[t] = 0.0f; }
  __syncthreads();
  if (t == 0) {
    // _first_k_true(distinct, 250): distinct positions in order, then the rest.
    int outp = 0;
    for (int i = 0; i < MRAND && outp < NRAND; ++i) {
      int a = rab[i], p = rpb[i], n = rnb[i];
      if ((a != p) && (a != n) && (p != n)) {
        r_a[outp] = a; r_p[outp] = p; r_n[outp] = n; r_mask[outp] = 1.0f; ++outp;
      }
    }
    for (int i = 0; i < MRAND && outp < NRAND; ++i) {
      int a = rab[i], p = rpb[i], n = rnb[i];
      if (!((a != p) && (a != n) && (p != n))) {
        r_a[outp] = a; r_p[outp] = p; r_n[outp] = n; r_mask[outp] = 0.0f; ++outp;
      }
    }
    for (int i = 0; i < NRAND; ++i) {
      int vp = val9(iabs_i(bidx[r_a[i]] - bidx[r_p[i]]));
      int vn = val9(iabs_i(bidx[r_a[i]] - bidx[r_n[i]]));
      r_vp[i] = (float)vp; r_vn[i] = (float)vn;
      if (!(vp > vn)) r_mask[i] = 0.0f;     // r_mask = r_valid & (vp > vn)
    }
  }
}

// ------------------------------------------- K1: gather za + row norms na,nz
__global__ void prep_kernel(const float* __restrict__ z, const int* __restrict__ anchors,
                            float* __restrict__ za, float* __restrict__ na,
                            float* __restrict__ nz) {
  __shared__ float red[64];
  int row = blockIdx.x;
  int t = threadIdx.x;
  float v;
  if (row < NCAND_PAD) {
    v = (row < NCAND) ? z[(size_t)anchors[row] * D_SZ + t] : 0.0f;
    za[(size_t)row * D_SZ + t] = v;
  } else {
    int j = row - NCAND_PAD;
    v = z[(size_t)j * D_SZ + t];
  }
  red[t] = v * v;
  __syncthreads();
  for (int s = 32; s > 0; s >>= 1) {
    if (t < s) red[t] += red[t + s];
    __syncthreads();
  }
  if (t == 0) {
    if (row < NCAND_PAD) na[row] = red[0];
    else                 nz[row - NCAND_PAD] = red[0];
  }
}

// ------------------------- K2: valuations + counting-sort ranks per anchor ci
__global__ void val_rank_kernel(const int* __restrict__ bidx, const int* __restrict__ anchors,
                                unsigned char* __restrict__ v_arr,
                                int* __restrict__ pos_idx, int* __restrict__ neg_idx) {
  __shared__ int scnt[256 * 10];
  __shared__ int hist[10];
  __shared__ int anchorRank;
  int ci = blockIdx.x;
  int t = threadIdx.x;
  int a = anchors[ci];
  int aval = bidx[a];
  unsigned char* vrow = v_arr + (size_t)ci * B_SZ;

  int j0 = t * 64, j1 = j0 + 64;
  int cnt[10];
#pragma unroll
  for (int c = 0; c < 10; ++c) cnt[c] = 0;
  for (int j = j0; j < j1; ++j) {
    int v = val9(iabs_i(aval - bidx[j]));
    vrow[j] = (unsigned char)v;
    cnt[v]++;
  }
#pragma unroll
  for (int c = 0; c < 10; ++c) scnt[t * 10 + c] = cnt[c];
  __syncthreads();
  // exclusive scan per class across the 256 thread-chunks (in j order)
  if (t < 10) {
    int run = 0;
    for (int tt = 0; tt < 256; ++tt) {
      int tmp = scnt[tt * 10 + t];
      scnt[tt * 10 + t] = run;
      run += tmp;
    }
    hist[t] = run;
  }
  __syncthreads();
  // class base offsets (descending valuation order)
  int classbase[10];
  {
    int run = 0;
    for (int c = 9; c >= 0; --c) { classbase[c] = run; run += hist[c]; }
  }
  // anchor's rank (anchor has v==9, classbase[9]==0)
  if (t == (a >> 6)) {
    int local = 0;
    for (int j = j0; j < a; ++j) if (vrow[j] == 9) ++local;
    anchorRank = scnt[t * 10 + 9] + local;
  }
  __syncthreads();
  int arank = anchorRank;
  int cnt2[10];
#pragma unroll
  for (int c = 0; c < 10; ++c) cnt2[c] = 0;
  for (int j = j0; j < j1; ++j) {
    int v = vrow[j];
    int rank = classbase[v] + scnt[t * 10 + v] + cnt2[v];
    cnt2[v]++;
    if (j == a) continue;                  // anchor removed from sorted list
    int r2 = rank - (rank > arank ? 1 : 0);
    if (r2 < NPOS)        pos_idx[ci * NPOS + r2] = j;
    else if (r2 >= NHALF) neg_idx[(size_t)ci * NNEG + (r2 - NHALF)] = j;
  }
}

// --------- K3: WMMA f32 GEMM (za @ z^T) -> d_rows = sqrt(clip(d2, 0))
__global__ void gemm_dist_kernel(const float* __restrict__ za, const float* __restrict__ z,
                                 const float* __restrict__ na, const float* __restrict__ nz,
                                 float* __restrict__ d_rows) {
  const int NT = B_SZ / 16;                       // 1024 N tiles
  int wave = threadIdx.x >> 5;
  int lane = threadIdx.x & 31;
  int tile = blockIdx.x * 4 + wave;
  int mt = tile / NT;                             // 0..62
  int nt = tile % NT;
  if (mt >= NCAND_PAD / 16) return;
  int m0 = mt * 16, n0 = nt * 16;
  int half = lane >> 4;                           // K sub-pair select
  int l = lane & 15;

  v8f acc = {0.f, 0.f, 0.f, 0.f, 0.f, 0.f, 0.f, 0.f};
  const float* arow = za + (size_t)(m0 + l) * D_SZ;   // A: M = l (both halves)
  const float* brow = z + (size_t)(n0 + l) * D_SZ;    // B[k][n] = z[n][k]: N = l
  for (int kt = 0; kt < D_SZ; kt += 4) {
    int kb = kt + half * 2;
    v2f a, b;
    a.x = arow[kb]; a.y = arow[kb + 1];
    b.x = brow[kb]; b.y = brow[kb + 1];
    if (kt + 4 < D_SZ) __builtin_prefetch((const void*)(brow + kt + 4), 0, 3);
    acc = __builtin_amdgcn_wmma_f32_16x16x4_f32(false, a, false, b,
                                                (short)0, acc, false, false);
  }
  int n = n0 + l;
  float nzv = nz[n];
#pragma unroll
  for (int r = 0; r < 8; ++r) {
    int m = m0 + r + 8 * half;                    // C/D layout: VGPR r -> M = r (+8 hi half)
    if (m < NCAND) {
      float d2 = (na[m] + nzv) - 2.0f * acc[r];
      d_rows[(size_t)m * B_SZ + n] = sqrtf(fmaxf(d2, 0.0f));
    }
  }
}

// --------------------------------- K4: count hard triplets per anchor ci
__global__ void hard_count_kernel(const float* __restrict__ d_rows,
                                  const unsigned char* __restrict__ v_arr,
                                  const int* __restrict__ pos_idx,
                                  const int* __restrict__ neg_idx,
                                  int* __restrict__ cnt) {
  __shared__ float dap[NPOS];
  __shared__ float vpv[NPOS];
  __shared__ int red[256];
  int ci = blockIdx.x;
  int t = threadIdx.x;
  const float* drow = d_rows + (size_t)ci * B_SZ;
  const unsigned char* vrow = v_arr + (size_t)ci * B_SZ;
  if (t < NPOS) {
    int p = pos_idx[ci * NPOS + t];
    dap[t] = drow[p];
    vpv[t] = (float)vrow[p];
  }
  __syncthreads();
  int local = 0;
  for (int ni = t; ni < NNEG; ni += 256) {
    int j = neg_idx[(size_t)ci * NNEG + ni];
    float dan = drow[j];
    float vn = (float)vrow[j];
#pragma unroll
    for (int pi = 0; pi < NPOS; ++pi)
      if (hard_cond(dan, dap[pi], vpv[pi], vn)) ++local;
  }
  red[t] = local;
  __syncthreads();
  for (int s = 128; s > 0; s >>= 1) {
    if (t < s) red[t] += red[t + s];
    __syncthreads();
  }
  if (t == 0) cnt[ci] = red[0];
}

// ----------------------- K5: exclusive scan over ci + hard validity mask
__global__ void scan_kernel(const int* __restrict__ cnt, int* __restrict__ base,
                            float* __restrict__ h_valid) {
  __shared__ int totalS;
  int t = threadIdx.x;
  if (t == 0) {
    int run = 0;
    for (int ci = 0; ci < NCAND; ++ci) { base[ci] = run; run += cnt[ci]; }
    totalS = run;
  }
  __syncthreads();
  int total = totalS;
  if (t < NHARD) h_valid[t] = (t < total) ? 1.0f : 0.0f;
}

// ---------------- K6: emit first-250 hard triplets in flattened order
__global__ void hard_emit_kernel(const float* __restrict__ d_rows,
                                 const unsigned char* __restrict__ v_arr,
                                 const int* __restrict__ pos_idx,
                                 const int* __restrict__ neg_idx,
                                 const int* __restrict__ anchors,
                                 const int* __restrict__ base,
                                 int* __restrict__ h_a, int* __restrict__ h_p,
                                 int* __restrict__ h_n,
                                 float* __restrict__ h_vp, float* __restrict__ h_vn) {
  __shared__ float dap[NPOS];
  __shared__ float vpv[NPOS];
  __shared__ int scn[256];
  int ci = blockIdx.x;
  int b0 = base[ci];
  if (b0 >= NHARD) return;
  int t = threadIdx.x;
  const float* drow = d_rows + (size_t)ci * B_SZ;
  const unsigned char* vrow = v_arr + (size_t)ci * B_SZ;
  if (t < NPOS) {
    int p = pos_idx[ci * NPOS + t];
    dap[t] = drow[p];
    vpv[t] = (float)vrow[p];
  }
  __syncthreads();
  // contiguous chunks over l = pi*NNEG + ni, 96 per thread (256*96 = 24576)
  int l0 = t * 96, l1 = l0 + 96;
  int local = 0;
  for (int l = l0; l < l1; ++l) {
    int pi = l / NNEG, ni = l % NNEG;
    int j = neg_idx[(size_t)ci * NNEG + ni];
    if (hard_cond(drow[j], dap[pi], vpv[pi], (float)vrow[j])) ++local;
  }
  scn[t] = local;
  __syncthreads();
  if (t == 0) {
    int run = 0;
    for (int tt = 0; tt < 256; ++tt) { int tmp = scn[tt]; scn[tt] = run; run += tmp; }
  }
  __syncthreads();
  int off = b0 + scn[t];
  if (off >= NHARD) return;                 // nothing from this chunk lands in first 250
  int emitted = 0;
  for (int l = l0; l < l1; ++l) {
    int pi = l / NNEG, ni = l % NNEG;
    int j = neg_idx[(size_t)ci * NNEG + ni];
    float vn = (float)vrow[j];
    if (hard_cond(drow[j], dap[pi], vpv[pi], vn)) {
      int pos = off + emitted;
      ++emitted;
      if (pos < NHARD) {
        h_a[pos] = anchors[ci];
        h_p[pos] = pos_idx[ci * NPOS + pi];
        h_n[pos] = j;
        h_vp[pos] = vpv[pi];
        h_vn[pos] = vn;
      }
    }
  }
}

// ----------------------------------------------- K7: final triplet loss
__global__ void loss_kernel(const float* __restrict__ z,
                            const int* __restrict__ h_a, const int* __restrict__ h_p,
                            const int* __restrict__ h_n,
                            const float* __restrict__ h_vp, const float* __restrict__ h_vn,
                            const float* __restrict__ h_valid,
                            const int* __restrict__ r_a, const int* __restrict__ r_p,
                            const int* __restrict__ r_n,
                            const float* __restrict__ r_vp, const float* __restrict__ r_vn,
                            const float* __restrict__ r_mask,
                            float* __restrict__ out) {
  __shared__ float rs[512];
  __shared__ float rm[512];
  int t = threadIdx.x;
  float contrib = 0.0f, mval = 0.0f;
  if (t < NTRIP) {
    int a, p, n; float vp, vn, mk;
    if (t < NHARD) {
      a = h_a[t]; p = h_p[t]; n = h_n[t];
      vp = h_vp[t]; vn = h_vn[t]; mk = h_valid[t];
    } else {
      int i = t - NHARD;
      a = r_a[i]; p = r_p[i]; n = r_n[i];
      vp = r_vp[i]; vn = r_vn[i]; mk = r_mask[i];
    }
    const float* zA = z + (size_t)a * D_SZ;
    const float* zP = z + (size_t)p * D_SZ;
    const float* zN = z + (size_t)n * D_SZ;
    float sp = 0.0f, sn = 0.0f;
    for (int k = 0; k < D_SZ; ++k) {
      float dp = zA[k] - zP[k];
      float dn = zA[k] - zN[k];
      sp += dp * dp;
      sn += dn * dn;
    }
    float d_pos = (sp > 0.0f) ? sqrtf(sp) : 0.0f;   // _safe_norm
    float d_neg = (sn > 0.0f) ? sqrtf(sn) : 0.0f;
    float margin = 0.05f + 0.15f * fabsf(vp - vn);
    float viol = d_pos - d_neg + margin;
    contrib = fmaxf(viol, 0.0f) * mk;
    mval = mk;
  }
  rs[t] = contrib;
  rm[t] = mval;
  __syncthreads();
  for (int s = 256; s > 0; s >>= 1) {
    if (t < s) { rs[t] += rs[t + s]; rm[t] += rm[t + s]; }
    __syncthreads();
  }
  if (t == 0) out[0] = rs[0] / fmaxf(rm[0], 1.0f);
}

// --------------------------------------------------------------- launcher
extern "C" void kernel_launch(void* const* d_in, const int* in_sizes, int n_in,
                              void* d_out, int out_size, void* d_ws, size_t ws_size,
                              hipStream_t stream) {
  (void)in_sizes; (void)n_in; (void)out_size; (void)ws_size;
  const float* z = (const float*)d_in[0];
  const int* bidx = (const int*)d_in[1];

  char* ws = (char*)d_ws;
  size_t off = 0;
  auto take = [&](size_t bytes) -> char* {
    char* p = ws + off;
    off = (off + bytes + 255) & ~(size_t)255;
    return p;
  };
  float*         d_rows  = (float*)take((size_t)NCAND * B_SZ * sizeof(float));
  float*         za      = (float*)take((size_t)NCAND_PAD * D_SZ * sizeof(float));
  float*         na      = (float*)take(NCAND_PAD * sizeof(float));
  float*         nz      = (float*)take(B_SZ * sizeof(float));
  unsigned char* v_arr   = (unsigned char*)take((size_t)NCAND * B_SZ);
  int*           neg_idx = (int*)take((size_t)NCAND * NNEG * sizeof(int));
  int*           pos_idx = (int*)take(NCAND * NPOS * sizeof(int));
  int*           anchors = (int*)take(NCAND * sizeof(int));
  int*           cnt     = (int*)take(NCAND * sizeof(int));
  int*           base    = (int*)take(NCAND * sizeof(int));
  int*           h_a     = (int*)take(NHARD * sizeof(int));
  int*           h_p     = (int*)take(NHARD * sizeof(int));
  int*           h_n     = (int*)take(NHARD * sizeof(int));
  float*         h_vp    = (float*)take(NHARD * sizeof(float));
  float*         h_vn    = (float*)take(NHARD * sizeof(float));
  float*         h_valid = (float*)take(NHARD * sizeof(float));
  int*           r_a     = (int*)take(NRAND * sizeof(int));
  int*           r_p     = (int*)take(NRAND * sizeof(int));
  int*           r_n     = (int*)take(NRAND * sizeof(int));
  float*         r_vp    = (float*)take(NRAND * sizeof(float));
  float*         r_vn    = (float*)take(NRAND * sizeof(float));
  float*         r_mask  = (float*)take(NRAND * sizeof(float));
  int*           rab     = (int*)take(MRAND * sizeof(int));
  int*           rpb     = (int*)take(MRAND * sizeof(int));
  int*           rnb     = (int*)take(MRAND * sizeof(int));

  rng_kernel<<<1, 512, 0, stream>>>(bidx, anchors, rab, rpb, rnb,
                                    r_a, r_p, r_n, r_vp, r_vn, r_mask,
                                    h_a, h_p, h_n, h_vp, h_vn);
  prep_kernel<<<NCAND_PAD + B_SZ, 64, 0, stream>>>(z, anchors, za, na, nz);
  val_rank_kernel<<<NCAND, 256, 0, stream>>>(bidx, anchors, v_arr, pos_idx, neg_idx);
  {
    int tiles = (NCAND_PAD / 16) * (B_SZ / 16);   // 63 * 1024
    gemm_dist_kernel<<<tiles / 4, 128, 0, stream>>>(za, z, na, nz, d_rows);
  }
  hard_count_kernel<<<NCAND, 256, 0, stream>>>(d_rows, v_arr, pos_idx, neg_idx, cnt);
  scan_kernel<<<1, 256, 0, stream>>>(cnt, base, h_valid);
  hard_emit_kernel<<<NCAND, 256, 0, stream>>>(d_rows, v_arr, pos_idx, neg_idx,
                                              anchors, base, h_a, h_p, h_n, h_vp, h_vn);
  loss_kernel<<<1, 512, 0, stream>>>(z, h_a, h_p, h_n, h_vp, h_vn, h_valid,
                                     r_a, r_p, r_n, r_vp, r_vn, r_mask,
                                     (float*)d_out);
}